// SparseAutoencoder_86354612454035
// MI455X (gfx1250) — compile-verified
//
#include <hip/hip_runtime.h>
#include <hip/hip_bf16.h>

#define D_IN   1024
#define D_HID  16384
#define BATCHN 8192
#define KSP    32

typedef __attribute__((ext_vector_type(2))) float v2f;
typedef __attribute__((ext_vector_type(8))) float v8f;
typedef int v4i_vs __attribute__((vector_size(16)));   // matches builtin param

// ---------------------------------------------------------------------------
// CDNA5 async global->LDS copy (ASYNCcnt-tracked). Falls back to a plain
// copy if the builtin is not available on this toolchain.
// ---------------------------------------------------------------------------
__device__ __forceinline__ void async_copy_b128(float4* lds_dst,
                                                const float4* gsrc)
{
#if __has_builtin(__builtin_amdgcn_global_load_async_to_lds_b128)
    __builtin_amdgcn_global_load_async_to_lds_b128(
        (__attribute__((address_space(1))) v4i_vs*)gsrc,
        (__attribute__((address_space(3))) v4i_vs*)lds_dst,
        /*offset=*/0, /*cpol=*/0);
#else
    *lds_dst = *gsrc;
#endif
}

__device__ __forceinline__ void async_copy_wait()
{
#if __has_builtin(__builtin_amdgcn_global_load_async_to_lds_b128)
    asm volatile("s_wait_asynccnt 0" ::: "memory");
#endif
}

// ---------------------------------------------------------------------------
// Encode: h[m, n] = sum_k x[m,k] * W_enc[n,k] + b_enc[n]
// Block: 256 threads (8 waves). Covers 16 M-rows x 1024 N-cols.
// Wave w covers N columns [n0 + w*128, n0 + w*128 + 128) via 8 WMMA tiles.
// Uses V_WMMA_F32_16X16X4_F32 (full fp32 precision).
// ---------------------------------------------------------------------------
__global__ __launch_bounds__(256) void sae_encode_kernel(
    const float* __restrict__ x, const float* __restrict__ W_enc,
    const float* __restrict__ b_enc, float* __restrict__ h)
{
    __shared__ float xs[16 * D_IN]; // 64 KB: 16 rows of x

    const int m0  = blockIdx.x * 16;
    const int n0  = blockIdx.y * 1024;
    const int tid = threadIdx.x;

    // Cooperative async stage of the x tile into LDS (coalesced b128)
    {
        const float4* xg  = (const float4*)(x + (size_t)m0 * D_IN);
        float4*       xs4 = (float4*)xs;
        #pragma unroll
        for (int i = 0; i < 16; ++i)
            async_copy_b128(&xs4[tid + i * 256], &xg[tid + i * 256]);
        async_copy_wait();
    }
    __syncthreads();

    const int wave  = tid >> 5;
    const int lane  = tid & 31;
    const int l16   = lane & 15;
    const int khalf = lane >> 4;          // 0: K=0,1  1: K=2,3 (f32 frag layout)
    const int nw    = n0 + wave * 128;

    v8f acc[8];
    #pragma unroll
    for (int i = 0; i < 8; ++i)
        #pragma unroll
        for (int j = 0; j < 8; ++j)
            acc[i][j] = 0.0f;

    // A fragment source (LDS): lane reads x[M=l16, K=k+2*khalf .. +1]
    const float* aptr  = xs + l16 * D_IN + 2 * khalf;
    // B fragment source (global, L2-resident): W_enc[N=nw+nt*16+l16, K=..]
    const float* bbase = W_enc + (size_t)(nw + l16) * D_IN + 2 * khalf;

    #pragma unroll 2
    for (int k = 0; k < D_IN; k += 4) {
        // keep the W_enc K-stream warm in WGP$ (global_prefetch_b8)
        __builtin_prefetch(bbase + k + 128, 0, 3);
        v2f a = *(const v2f*)(aptr + k);
        #pragma unroll
        for (int nt = 0; nt < 8; ++nt) {
            v2f b = *(const v2f*)(bbase + (size_t)(nt * 16) * D_IN + k);
            acc[nt] = __builtin_amdgcn_wmma_f32_16x16x4_f32(
                /*neg_a=*/false, a, /*neg_b=*/false, b,
                /*c_mod=*/(short)0, acc[nt],
                /*reuse_a=*/false, /*reuse_b=*/false);
        }
    }

    // D tile: VGPR v holds (M = v + 8*khalf, N = l16) per half-wave.
    const int mbase = m0 + 8 * khalf;
    #pragma unroll
    for (int nt = 0; nt < 8; ++nt) {
        const int   n    = nw + nt * 16 + l16;
        const float bias = b_enc[n];
        #pragma unroll
        for (int v = 0; v < 8; ++v) {
            h[(size_t)(mbase + v) * D_HID + n] = acc[nt][v] + bias;
        }
    }
}

// ---------------------------------------------------------------------------
// Top-K: one block (256 threads) per row. 32 rounds of block-wide argmax
// over an LDS-staged row; strict '>' keeps lowest index on ties within a
// thread's scan; the tree reduction need not. Rewrites the row masked; saves (idx,val) pairs for decode.
// ---------------------------------------------------------------------------
__global__ __launch_bounds__(256) void sae_topk_kernel(
    float* __restrict__ h, int* __restrict__ sel_idx_g,
    float* __restrict__ sel_val_g)
{
    __shared__ float sh[D_HID];   // 64 KB row stage
    __shared__ float rmax[256];
    __shared__ int   ridx[256];
    __shared__ int   sidx[KSP];
    __shared__ float sval[KSP];

    const int row = blockIdx.x;
    const int tid = threadIdx.x;
    float* hrow   = h + (size_t)row * D_HID;

    {
        const float4* hg  = (const float4*)hrow;
        float4*       sh4 = (float4*)sh;
        #pragma unroll
        for (int i = 0; i < 16; ++i)
            async_copy_b128(&sh4[tid + i * 256], &hg[tid + i * 256]);
        async_copy_wait();
    }
    __syncthreads();

    const float NEG = -__builtin_inff();
    const int base  = tid * 64;

    for (int it = 0; it < KSP; ++it) {
        float m  = NEG;
        int   mi = base;
        for (int j = 0; j < 64; ++j) {
            float v = sh[base + j];
            if (v > m) { m = v; mi = base + j; }
        }
        rmax[tid] = m;
        ridx[tid] = mi;
        __syncthreads();
        for (int s = 128; s > 0; s >>= 1) {
            if (tid < s) {
                if (rmax[tid + s] > rmax[tid]) {
                    rmax[tid] = rmax[tid + s];
                    ridx[tid] = ridx[tid + s];
                }
            }
            __syncthreads();
        }
        if (tid == 0) {
            int ix   = ridx[0];
            sidx[it] = ix;
            sval[it] = rmax[0];
            sh[ix]   = NEG;       // remove from further consideration
        }
        __syncthreads();
    }

    // Rewrite masked row: zeros everywhere except the selected 32.
    for (int j = 0; j < 64; ++j) {
        if (sh[base + j] != NEG) hrow[base + j] = 0.0f;
    }
    if (tid < KSP) {
        const int   ix = sidx[tid];
        const float v  = sval[tid];
        hrow[ix] = v;
        sel_idx_g[row * KSP + tid] = ix;
        sel_val_g[row * KSP + tid] = v;
    }
}

// ---------------------------------------------------------------------------
// Decode: x_hat[b, o] = sum_j sval[j] * W_dec[o, sidx[j]] + b_dec[o]
// One block per row; W_dec column gathers hit L2 (W_dec = 64 MB < 192 MB).
// ---------------------------------------------------------------------------
__global__ __launch_bounds__(256) void sae_decode_kernel(
    const float* __restrict__ W_dec, const float* __restrict__ b_dec,
    const int* __restrict__ sel_idx_g, const float* __restrict__ sel_val_g,
    float* __restrict__ xhat)
{
    __shared__ int   sidx[KSP];
    __shared__ float sval[KSP];

    const int row = blockIdx.x;
    const int tid = threadIdx.x;

    if (tid < KSP) {
        sidx[tid] = sel_idx_g[row * KSP + tid];
        sval[tid] = sel_val_g[row * KSP + tid];
    }
    __syncthreads();

    float a0 = b_dec[tid];
    float a1 = b_dec[tid + 256];
    float a2 = b_dec[tid + 512];
    float a3 = b_dec[tid + 768];

    #pragma unroll 4
    for (int j = 0; j < KSP; ++j) {
        const int   ix = sidx[j];
        const float v  = sval[j];
        a0 += v * W_dec[(size_t)(tid      ) * D_HID + ix];
        a1 += v * W_dec[(size_t)(tid + 256) * D_HID + ix];
        a2 += v * W_dec[(size_t)(tid + 512) * D_HID + ix];
        a3 += v * W_dec[(size_t)(tid + 768) * D_HID + ix];
    }

    float* out = xhat + (size_t)row * D_IN;
    out[tid      ] = a0;
    out[tid + 256] = a1;
    out[tid + 512] = a2;
    out[tid + 768] = a3;
}

// ---------------------------------------------------------------------------
extern "C" void kernel_launch(void* const* d_in, const int* in_sizes, int n_in,
                              void* d_out, int out_size, void* d_ws, size_t ws_size,
                              hipStream_t stream)
{
    const float* x     = (const float*)d_in[0];
    const float* W_enc = (const float*)d_in[1];
    const float* b_enc = (const float*)d_in[2];
    const float* W_dec = (const float*)d_in[3];
    const float* b_dec = (const float*)d_in[4];

    float* xhat = (float*)d_out;                              // [8192, 1024]
    float* h    = (float*)d_out + (size_t)BATCHN * D_IN;      // [8192, 16384]

    int*   sel_idx = (int*)d_ws;                              // [8192, 32]
    float* sel_val = (float*)((char*)d_ws +
                              sizeof(int) * (size_t)BATCHN * KSP);

    dim3 gridE(BATCHN / 16, D_HID / 1024);
    sae_encode_kernel<<<gridE, 256, 0, stream>>>(x, W_enc, b_enc, h);
    sae_topk_kernel<<<BATCHN, 256, 0, stream>>>(h, sel_idx, sel_val);
    sae_decode_kernel<<<BATCHN, 256, 0, stream>>>(W_dec, b_dec,
                                                  sel_idx, sel_val, xhat);
}